// GCNClassification_57140244906592
// MI455X (gfx1250) — compile-verified
//
#include <hip/hip_runtime.h>
#include <hip/hip_bf16.h>
#include <math.h>

#define N_NODES   100000
#define N_EDGES   1600000
#define HEADS     4
#define CH        32
#define HC        (HEADS*CH)   /* 128 */
#define FIN       7
#define KPAD      8            /* K padded to 8 -> two K=4 WMMAs */
#define N_IDS     10000
#define NEG_SLOPE 0.2f
#define NTILES    (N_NODES / 16)   /* 6250, exact */
#define TILES_PER_BLOCK 8          /* 8 waves of 32 per block */

typedef float v2f __attribute__((ext_vector_type(2)));
typedef float v8f __attribute__((ext_vector_type(8)));

// Monotonic float -> uint encoding so float max == uint max.
__device__ __forceinline__ unsigned enc_f32(float f) {
    unsigned b = __float_as_uint(f);
    return (b & 0x80000000u) ? ~b : (b | 0x80000000u);
}
__device__ __forceinline__ float dec_f32(unsigned u) {
    unsigned b = (u & 0x80000000u) ? (u ^ 0x80000000u) : ~u;
    return __uint_as_float(b);
}
__device__ __forceinline__ float lrelu(float v) {
    return v > 0.f ? v : NEG_SLOPE * v;
}

// ---------------------------------------------------------------------------
// K0: initialize scratch accumulators (deterministic: runs every launch)
// ---------------------------------------------------------------------------
__global__ void init_ws_kernel(float* __restrict__ out_acc,
                               float* __restrict__ denom,
                               unsigned* __restrict__ amax) {
    int stride = gridDim.x * blockDim.x;
    int tid = blockIdx.x * blockDim.x + threadIdx.x;
    for (int j = tid; j < N_NODES * HC; j += stride) out_acc[j] = 0.f;
    for (int j = tid; j < N_NODES * HEADS; j += stride) {
        denom[j] = 0.f;
        amax[j]  = 0x007FFFFFu;   // enc(-inf)
    }
}

// ---------------------------------------------------------------------------
// K1: h = x @ W  via V_WMMA_F32_16X16X4_F32.
// Zero-padded operands are staged in LDS (branch-free clamped loads), so all
// fragment reads are unconditional ds loads and EXEC stays all-ones around
// every WMMA. One wave computes a 16-node x 128-col stripe (8 N-tiles).
// ---------------------------------------------------------------------------
__global__ void wmma_proj_kernel(const float* __restrict__ x,
                                 const float* __restrict__ W,
                                 float* __restrict__ h) {
    __shared__ float sWT[HC * KPAD];                      // W transposed+padded: [n][k]
    __shared__ float sX[TILES_PER_BLOCK * 16 * KPAD];     // x tiles padded: [tile][row][k]

    int tid       = threadIdx.x;
    int waveInBlk = tid >> 5;
    int lane      = tid & 31;
    int tileBase  = blockIdx.x * TILES_PER_BLOCK;

    // Stage W^T zero-padded to K=8 (branch-free: clamp index, select 0)
    for (int idx = tid; idx < HC * KPAD; idx += blockDim.x) {
        int n = idx >> 3, k = idx & 7;
        float v = W[min(k, FIN - 1) * HC + n];
        sWT[idx] = (k < FIN) ? v : 0.f;
    }
    // Stage 8 x-tiles zero-padded to K=8 (clamp row for tail tiles)
    for (int idx = tid; idx < TILES_PER_BLOCK * 16 * KPAD; idx += blockDim.x) {
        int t = idx >> 7, r = (idx >> 3) & 15, k = idx & 7;
        int m = (tileBase + t) * 16 + r;
        int mc = min(m, N_NODES - 1);
        float v = x[(size_t)mc * FIN + min(k, FIN - 1)];
        sX[idx] = (k < FIN) ? v : 0.f;
    }
    __syncthreads();

    int tile = tileBase + waveInBlk;
    if (tile < NTILES) {
        int koff = (lane >= 16) ? 2 : 0;   // A 16x4 layout: hi half-wave holds K2/K3
        const float* xrow = &sX[(waveInBlk * 16 + (lane & 15)) * KPAD];

        v2f a0, a1;                         // contiguous in LDS -> ds_load_b64
        a0.x = xrow[koff];     a0.y = xrow[koff + 1];
        a1.x = xrow[4 + koff]; a1.y = xrow[4 + koff + 1];

        int ncol  = lane & 15;
        int mbase = tile * 16 + ((lane < 16) ? 0 : 8);
#pragma unroll
        for (int t = 0; t < 8; ++t) {
            int n = t * 16 + ncol;
            const float* wrow = &sWT[n * KPAD];
            v2f b0, b1;
            b0.x = wrow[koff];     b0.y = wrow[koff + 1];
            b1.x = wrow[4 + koff]; b1.y = wrow[4 + koff + 1];

            v8f acc = {};
            acc = __builtin_amdgcn_wmma_f32_16x16x4_f32(false, a0, false, b0,
                                                        (short)0, acc, false, false);
            acc = __builtin_amdgcn_wmma_f32_16x16x4_f32(false, a1, false, b1,
                                                        (short)0, acc, false, false);
#pragma unroll
            for (int r = 0; r < 8; ++r) {
                h[(size_t)(mbase + r) * HC + n] = acc[r];   // tiles are always full
            }
        }
    }
}

// ---------------------------------------------------------------------------
// K2: per-(node,head) attention coefficients a_src, a_dst
// ---------------------------------------------------------------------------
__global__ void attn_coef_kernel(const float* __restrict__ h,
                                 const float* __restrict__ att_src,
                                 const float* __restrict__ att_dst,
                                 float* __restrict__ a_src,
                                 float* __restrict__ a_dst) {
    int i = blockIdx.x * blockDim.x + threadIdx.x;
    if (i >= N_NODES * HEADS) return;
    int hd = i & (HEADS - 1);
    int n  = i >> 2;
    const float* hr = h + (size_t)n * HC + hd * CH;
    const float* as = att_src + hd * CH;
    const float* ad = att_dst + hd * CH;
    float s1 = 0.f, s2 = 0.f;
#pragma unroll
    for (int c = 0; c < CH; ++c) {
        float v = hr[c];
        s1 += v * as[c];
        s2 += v * ad[c];
    }
    a_src[i] = s1;
    a_dst[i] = s2;
}

// ---------------------------------------------------------------------------
// K3: segment max over incoming edges (encoded-uint atomicMax)
// ---------------------------------------------------------------------------
__global__ void edge_max_kernel(const int* __restrict__ ei,
                                const float* __restrict__ a_src,
                                const float* __restrict__ a_dst,
                                unsigned* __restrict__ amax) {
    int i = blockIdx.x * blockDim.x + threadIdx.x;
    if (i >= N_EDGES * HEADS) return;
    int e  = i >> 2;
    int hd = i & 3;
    int s = ei[e];
    int d = ei[N_EDGES + e];
    float al = lrelu(a_src[s * HEADS + hd] + a_dst[d * HEADS + hd]);
    atomicMax(&amax[d * HEADS + hd], enc_f32(al));
}

// ---------------------------------------------------------------------------
// K4: segment sum of exp(alpha - amax)
// ---------------------------------------------------------------------------
__global__ void edge_denom_kernel(const int* __restrict__ ei,
                                  const float* __restrict__ a_src,
                                  const float* __restrict__ a_dst,
                                  const unsigned* __restrict__ amax,
                                  float* __restrict__ denom) {
    int i = blockIdx.x * blockDim.x + threadIdx.x;
    if (i >= N_EDGES * HEADS) return;
    int e  = i >> 2;
    int hd = i & 3;
    int s = ei[e];
    int d = ei[N_EDGES + e];
    float al = lrelu(a_src[s * HEADS + hd] + a_dst[d * HEADS + hd]);
    float m  = dec_f32(amax[d * HEADS + hd]);
    atomicAdd(&denom[d * HEADS + hd], expf(al - m));
}

// ---------------------------------------------------------------------------
// K5: normalized weighted scatter-add of h[src] into out_acc[dst]
// ---------------------------------------------------------------------------
__global__ void edge_scatter_kernel(const int* __restrict__ ei,
                                    const float* __restrict__ a_src,
                                    const float* __restrict__ a_dst,
                                    const unsigned* __restrict__ amax,
                                    const float* __restrict__ denom,
                                    const float* __restrict__ h,
                                    float* __restrict__ out_acc) {
    int i = blockIdx.x * blockDim.x + threadIdx.x;
    if (i >= N_EDGES * HEADS) return;
    int e  = i >> 2;
    int hd = i & 3;
    int s = ei[e];
    int d = ei[N_EDGES + e];
    float al = lrelu(a_src[s * HEADS + hd] + a_dst[d * HEADS + hd]);
    float m  = dec_f32(amax[d * HEADS + hd]);
    float w  = expf(al - m) / (denom[d * HEADS + hd] + 1e-16f);

    const float4* hs = (const float4*)(h + (size_t)s * HC + hd * CH);
    float* od = out_acc + (size_t)d * HC + hd * CH;
#pragma unroll
    for (int q = 0; q < CH / 4; ++q) {
        float4 v = hs[q];
        atomicAdd(od + q * 4 + 0, v.x * w);
        atomicAdd(od + q * 4 + 1, v.y * w);
        atomicAdd(od + q * 4 + 2, v.z * w);
        atomicAdd(od + q * 4 + 3, v.w * w);
    }
}

// ---------------------------------------------------------------------------
// K6: only the queried ids: +bias, LayerNorm, classifier, softmax
// ---------------------------------------------------------------------------
__global__ void finalize_ids_kernel(const int* __restrict__ ids,
                                    const float* __restrict__ out_acc,
                                    const float* __restrict__ gat_bias,
                                    const float* __restrict__ ln_w,
                                    const float* __restrict__ ln_b,
                                    const float* __restrict__ lin_W,
                                    const float* __restrict__ lin_b,
                                    float* __restrict__ probs) {
    int i = blockIdx.x * blockDim.x + threadIdx.x;
    if (i >= N_IDS) return;
    int n = ids[i];
    const float* row = out_acc + (size_t)n * HC;

    float sum = 0.f, sumsq = 0.f;
    for (int j = 0; j < HC; ++j) {
        float v = row[j] + gat_bias[j];
        sum   += v;
        sumsq += v * v;
    }
    float mu  = sum * (1.f / HC);
    float var = sumsq * (1.f / HC) - mu * mu;
    float inv = rsqrtf(var + 1e-5f);

    float logit[FIN];
#pragma unroll
    for (int k = 0; k < FIN; ++k) logit[k] = lin_b[k];
    for (int j = 0; j < HC; ++j) {
        float v  = row[j] + gat_bias[j];
        float xn = (v - mu) * inv * ln_w[j] + ln_b[j];
#pragma unroll
        for (int k = 0; k < FIN; ++k) logit[k] += xn * lin_W[j * FIN + k];
    }
    float mx = logit[0];
#pragma unroll
    for (int k = 1; k < FIN; ++k) mx = fmaxf(mx, logit[k]);
    float se = 0.f, ex[FIN];
#pragma unroll
    for (int k = 0; k < FIN; ++k) { ex[k] = expf(logit[k] - mx); se += ex[k]; }
    float rse = 1.f / se;
#pragma unroll
    for (int k = 0; k < FIN; ++k) probs[(size_t)i * FIN + k] = ex[k] * rse;
}

// ---------------------------------------------------------------------------
extern "C" void kernel_launch(void* const* d_in, const int* in_sizes, int n_in,
                              void* d_out, int out_size, void* d_ws, size_t ws_size,
                              hipStream_t stream) {
    const float* x        = (const float*)d_in[0];
    /* d_in[1] = edge_weight: unused by GATConv (edge_dim=None) */
    const float* W        = (const float*)d_in[2];
    const float* att_src  = (const float*)d_in[3];
    const float* att_dst  = (const float*)d_in[4];
    const float* gat_bias = (const float*)d_in[5];
    const float* ln_w     = (const float*)d_in[6];
    const float* ln_b     = (const float*)d_in[7];
    const float* lin_W    = (const float*)d_in[8];
    const float* lin_b    = (const float*)d_in[9];
    const int*   ei       = (const int*)d_in[10];
    const int*   ids      = (const int*)d_in[11];
    float*       probs    = (float*)d_out;

    // Workspace layout:
    //   h        : N*128 f32  (51.2 MB)  projected features (L2-resident)
    //   out_acc  : N*128 f32  (51.2 MB)  scatter accumulator (L2-resident)
    //   a_src    : N*4 f32
    //   a_dst    : N*4 f32
    //   denom    : N*4 f32
    //   amax     : N*4 u32 (monotonic-encoded floats)
    char* ws = (char*)d_ws;
    float*    h       = (float*)ws;  ws += (size_t)N_NODES * HC    * sizeof(float);
    float*    out_acc = (float*)ws;  ws += (size_t)N_NODES * HC    * sizeof(float);
    float*    a_src   = (float*)ws;  ws += (size_t)N_NODES * HEADS * sizeof(float);
    float*    a_dst   = (float*)ws;  ws += (size_t)N_NODES * HEADS * sizeof(float);
    float*    denom   = (float*)ws;  ws += (size_t)N_NODES * HEADS * sizeof(float);
    unsigned* amax    = (unsigned*)ws;

    init_ws_kernel<<<2048, 256, 0, stream>>>(out_acc, denom, amax);

    // 8 waves per 256-thread block, one 16-node tile per wave
    int proj_blocks = (NTILES + TILES_PER_BLOCK - 1) / TILES_PER_BLOCK;
    wmma_proj_kernel<<<proj_blocks, 256, 0, stream>>>(x, W, h);

    attn_coef_kernel<<<(N_NODES * HEADS + 255) / 256, 256, 0, stream>>>(
        h, att_src, att_dst, a_src, a_dst);

    int edge_blocks = (N_EDGES * HEADS + 255) / 256;
    edge_max_kernel<<<edge_blocks, 256, 0, stream>>>(ei, a_src, a_dst, amax);
    edge_denom_kernel<<<edge_blocks, 256, 0, stream>>>(ei, a_src, a_dst, amax, denom);
    edge_scatter_kernel<<<edge_blocks, 256, 0, stream>>>(ei, a_src, a_dst, amax,
                                                         denom, h, out_acc);

    finalize_ids_kernel<<<(N_IDS + 255) / 256, 256, 0, stream>>>(
        ids, out_acc, gat_bias, ln_w, ln_b, lin_W, lin_b, probs);
}